// Loss_60430189855357
// MI455X (gfx1250) — compile-verified
//
#include <hip/hip_runtime.h>
#include <hip/hip_bf16.h>

typedef float v2f __attribute__((ext_vector_type(2)));
typedef float v8f __attribute__((ext_vector_type(8)));

#define BATCH 32
#define SEQ 4
#define TLEN 4000
#define FEAT 96
#define TILES_PER_B (TLEN / 16)          // 250
#define NWAVES (BATCH * TILES_PER_B)     // 8000
#define WAVES_PER_BLOCK 8
#define NBLOCKS (NWAVES / WAVES_PER_BLOCK) // 1000
#define BT_F 128000.0f

// Kernel 1: one wave32 computes 16 frames (b, t0..t0+15).
// For each s in 0..3 it computes 16 logits with 24 chained
// V_WMMA_F32_16X16X4_F32 ops (A = 16x4 chunk of x rows, B = W chunk
// broadcast over all 16 columns), then accumulates BCE and the
// per-frame match / label-zero / pred-zero flags across s.
__global__ __launch_bounds__(256) void
loss_wmma_kernel(const float* __restrict__ x,
                 const float* __restrict__ labels,
                 const float* __restrict__ W,
                 const float* __restrict__ bias_p,
                 float* __restrict__ bce_partial,
                 unsigned* __restrict__ cnt_partial)
{
    __shared__ float sW[FEAT];
    __shared__ float sLog[WAVES_PER_BLOCK][16];

    const int tid = threadIdx.x;
    if (tid < FEAT) sW[tid] = W[tid];
    __syncthreads();

    const int lane  = tid & 31;
    const int wslot = tid >> 5;
    const int wid   = blockIdx.x * WAVES_PER_BLOCK + wslot;  // 0..7999, exact grid

    const int b  = wid / TILES_PER_B;
    const int t0 = (wid - b * TILES_PER_B) * 16;

    const float bias = bias_p[0];
    const int m    = lane & 15;          // row within 16-frame tile
    const int koff = (lane >> 4) * 2;    // K pair: lanes 0-15 -> {0,1}, 16-31 -> {2,3}

    float bce = 0.0f;
    bool matchF = true, labzero = true, predzero = true;

    for (int s = 0; s < SEQ; ++s) {
        const int row0 = ((b * SEQ + s) * TLEN + t0);        // first row of tile
        const float* rowp = x + (size_t)row0 * FEAT + m * FEAT + koff;

        v8f c = {0.f, 0.f, 0.f, 0.f, 0.f, 0.f, 0.f, 0.f};
        #pragma unroll
        for (int ch = 0; ch < FEAT / 4; ++ch) {
            // A tile: lane (m, koff) -> x[row m, 4*ch + koff .. +1], streamed (no reuse)
            v2f a = __builtin_nontemporal_load((const v2f*)(rowp + ch * 4));
            // B tile: W chunk replicated across all 16 columns
            v2f bb;
            bb.x = sW[ch * 4 + koff];
            bb.y = sW[ch * 4 + koff + 1];
            c = __builtin_amdgcn_wmma_f32_16x16x4_f32(
                    /*neg_a=*/false, a, /*neg_b=*/false, bb,
                    /*c_mod=*/(short)0, c, /*reuse_a=*/false, /*reuse_b=*/false);
        }

        // D[m,n] identical over n: lanes 0-15 hold M=0..7 in c[0..7],
        // lanes 16-31 hold M=8..15. Redistribute via LDS: one logit per lane.
        if (lane == 0) {
            #pragma unroll
            for (int r = 0; r < 8; ++r) sLog[wslot][r] = c[r];
        }
        if (lane == 16) {
            #pragma unroll
            for (int r = 0; r < 8; ++r) sLog[wslot][8 + r] = c[r];
        }
        // DS ops from the same wave execute in order; read sees the stores.
        const float z = sLog[wslot][m] + bias;
        const float y = labels[(size_t)((b * SEQ + s) * TLEN) + t0 + m];

        if (lane < 16) {
            // stable softplus(z) - z*y
            bce += log1pf(expf(-fabsf(z))) + fmaxf(z, 0.0f) - z * y;
            const bool pred = z > 0.0f;
            matchF   = matchF   && ((y != 0.0f) == pred);
            labzero  = labzero  && (y == 0.0f);
            predzero = predzero && !pred;
        }
    }

    const bool valid = lane < 16;
    const unsigned long long mC  = __ballot(valid && matchF);
    const unsigned long long mFA = __ballot(valid && !matchF && labzero);
    const unsigned long long mMS = __ballot(valid && !matchF && !labzero && predzero);

    float v = bce;  // lanes >= 16 contribute 0
    #pragma unroll
    for (int off = 16; off > 0; off >>= 1) v += __shfl_down(v, off, 32);

    if (lane == 0) {
        bce_partial[wid] = v;
        cnt_partial[wid] = (unsigned)__popcll(mC)
                         | ((unsigned)__popcll(mFA) << 8)
                         | ((unsigned)__popcll(mMS) << 16);
    }
}

// Kernel 2: deterministic fixed-order reduction of the 8000 partials,
// then reproduce the reference's sequential metric normalization exactly.
__global__ __launch_bounds__(256) void
finalize_kernel(const float* __restrict__ bce_partial,
                const unsigned* __restrict__ cnt_partial,
                float* __restrict__ out)
{
    __shared__ float    sb[256];
    __shared__ unsigned s0[256], s1[256], s2[256];
    const int tid = threadIdx.x;

    float acc = 0.0f;
    unsigned c0 = 0, c1 = 0, c2 = 0;
    for (int i = tid; i < NWAVES; i += 256) {
        acc += bce_partial[i];
        const unsigned p = cnt_partial[i];
        c0 += p & 0xFFu;
        c1 += (p >> 8) & 0xFFu;
        c2 += (p >> 16) & 0xFFu;
    }
    sb[tid] = acc; s0[tid] = c0; s1[tid] = c1; s2[tid] = c2;
    __syncthreads();
    for (int st = 128; st > 0; st >>= 1) {
        if (tid < st) {
            sb[tid] += sb[tid + st];
            s0[tid] += s0[tid + st];
            s1[tid] += s1[tid + st];
            s2[tid] += s2[tid + st];
        }
        __syncthreads();
    }

    if (tid == 0) {
        const float Ssum = sb[0];
        const float total_loss = Ssum / BT_F + Ssum / 4.0f;
        const float loss = total_loss / BT_F;

        float correct = (float)s0[0];
        float FA = (float)s1[0];
        float MS = (float)s2[0];
        float SC = BT_F - correct - FA - MS;

        const float DER = (MS + FA + SC) / (MS + FA + SC + correct);
        MS = MS / (MS + FA + SC + correct);     // denominators mutate sequentially,
        FA = FA / (MS + FA + SC + correct);     // exactly as in the reference
        SC = SC / (MS + FA + SC + correct);

        out[0] = loss;
        out[1] = DER;
        out[2] = MS;
        out[3] = FA;
        out[4] = SC;
    }
}

extern "C" void kernel_launch(void* const* d_in, const int* in_sizes, int n_in,
                              void* d_out, int out_size, void* d_ws, size_t ws_size,
                              hipStream_t stream) {
    const float* x      = (const float*)d_in[0];   // [32,4,4000,96] f32
    const float* labels = (const float*)d_in[1];   // [32,4,4000]    f32
    const float* W      = (const float*)d_in[2];   // [1,96]         f32
    const float* bias   = (const float*)d_in[3];   // [1]            f32

    float*    bce_partial = (float*)d_ws;                               // 8000 f32
    unsigned* cnt_partial = (unsigned*)((char*)d_ws + NWAVES * sizeof(float)); // 8000 u32

    loss_wmma_kernel<<<NBLOCKS, 256, 0, stream>>>(x, labels, W, bias,
                                                  bce_partial, cnt_partial);
    finalize_kernel<<<1, 256, 0, stream>>>(bce_partial, cnt_partial, (float*)d_out);
}